// OTLoss_45226005627448
// MI455X (gfx1250) — compile-verified
//
#include <hip/hip_runtime.h>
#include <hip/hip_bf16.h>

// Sinkhorn OT loss, MI455X (gfx1250, wave32).
//   - distance tiles via V_WMMA_F32_16X16X4_F32 (fp32 WMMA, 16 per 16x16 tile)
//   - median via 2-level 4096-bin histogram (tiles recomputed, no 256MB M array)
//   - K and K^T materialized in fp16 (128MB each -> L2-friendly GEMV iterations)
//   - loss recomputes d2 in fp32 (avoids ln(fp16) / 0*inf issues)

#define NPTS  8192
#define DIMS  64
#define TILES 512            // NPTS / 16
#define NBINS 4096
#define INV_REG 10.0f        // 1 / SINKHORN_REG
#define NITER 50

typedef __attribute__((ext_vector_type(2))) float    v2f;
typedef __attribute__((ext_vector_type(8))) float    v8f;
typedef __attribute__((ext_vector_type(8))) _Float16 h8;

struct OTParams {
  unsigned minBits;
  unsigned maxBits;
  float lo2, w2;
  unsigned long long below2;
  float med, kscale, mscale;
  unsigned hist1[NBINS];
  unsigned hist2[NBINS];
};

__device__ __forceinline__ float wave_sum(float x) {
#pragma unroll
  for (int o = 16; o > 0; o >>= 1) x += __shfl_xor(x, o, 32);
  return x;
}
__device__ __forceinline__ float wave_min(float x) {
#pragma unroll
  for (int o = 16; o > 0; o >>= 1) x = fminf(x, __shfl_xor(x, o, 32));
  return x;
}
__device__ __forceinline__ float wave_max(float x) {
#pragma unroll
  for (int o = 16; o > 0; o >>= 1) x = fmaxf(x, __shfl_xor(x, o, 32));
  return x;
}

// One 16x16 tile of X @ Y^T (K-dim = 64) with V_WMMA_F32_16X16X4_F32.
// A layout: lanes 0-15 hold row=lane, K={0,1}; lanes 16-31 row=lane-16, K={2,3}.
// B layout mirrors with Y rows as columns. Result c[r]: row = row0+r+8*hl,
// col = col0+(lane&15).
__device__ __forceinline__ v8f tile_dot(const float* __restrict__ X,
                                        const float* __restrict__ Y,
                                        int row0, int col0, int lid, int koff) {
  v8f c = {0.f, 0.f, 0.f, 0.f, 0.f, 0.f, 0.f, 0.f};
  const float* xr = X + (size_t)(row0 + lid) * DIMS + koff;
  const float* yr = Y + (size_t)(col0 + lid) * DIMS + koff;
#pragma unroll
  for (int k = 0; k < DIMS; k += 4) {
    v2f a = {xr[k], xr[k + 1]};
    v2f b = {yr[k], yr[k + 1]};
    c = __builtin_amdgcn_wmma_f32_16x16x4_f32(false, a, false, b, (short)0, c,
                                              false, false);
  }
  return c;
}

__global__ void ot_init(OTParams* p, float* u, float* v, float* out) {
  int idx = blockIdx.x * blockDim.x + threadIdx.x;   // 8192 threads total
  if (idx < NBINS) { p->hist1[idx] = 0; p->hist2[idx] = 0; }
  if (idx < NPTS)  { u[idx] = 1.0f / NPTS; v[idx] = 1.0f / NPTS; }
  if (idx == 0) {
    p->minBits = 0x7f800000u;   // +inf
    p->maxBits = 0u;
    out[0] = 0.f;
  }
}

__global__ __launch_bounds__(256) void ot_rownorm(const float* __restrict__ xp,
                                                  const float* __restrict__ xt,
                                                  float* __restrict__ x2,
                                                  float* __restrict__ y2) {
  int lane = threadIdx.x & 31, wv = threadIdx.x >> 5;
  int row = blockIdx.x * 8 + wv;                     // 0 .. 2*NPTS-1
  const float* src = (row < NPTS) ? (xp + (size_t)row * DIMS)
                                  : (xt + (size_t)(row - NPTS) * DIMS);
  float a = src[lane * 2], b = src[lane * 2 + 1];
  float s = wave_sum(a * a + b * b);
  if (lane == 0) {
    if (row < NPTS) x2[row] = s; else y2[row - NPTS] = s;
  }
}

__global__ __launch_bounds__(256) void ot_minmax(const float* __restrict__ xp,
                                                 const float* __restrict__ xt,
                                                 const float* __restrict__ x2,
                                                 const float* __restrict__ y2,
                                                 OTParams* p) {
  int lane = threadIdx.x & 31, wv = threadIdx.x >> 5;
  int tile = blockIdx.x * 8 + wv;
  int ti = tile >> 9, tj = tile & (TILES - 1);
  int lid = lane & 15, hl = lane >> 4, koff = hl * 2;
  v8f c = tile_dot(xp, xt, ti * 16, tj * 16, lid, koff);
  int j = tj * 16 + lid;
  float yj = y2[j];
  float mn = 3.4e38f, mx = 0.f;
#pragma unroll
  for (int r = 0; r < 8; ++r) {
    int i = ti * 16 + r + 8 * hl;
    float d2 = fmaxf(x2[i] + yj - 2.0f * c[r], 0.0f);
    mn = fminf(mn, d2);
    mx = fmaxf(mx, d2);
  }
  mn = wave_min(mn);
  mx = wave_max(mx);
  if (lane == 0) {                       // d2 >= 0 -> float bits are monotone
    atomicMin(&p->minBits, __float_as_uint(mn));
    atomicMax(&p->maxBits, __float_as_uint(mx));
  }
}

__global__ __launch_bounds__(256) void ot_hist(const float* __restrict__ xp,
                                               const float* __restrict__ xt,
                                               const float* __restrict__ x2,
                                               const float* __restrict__ y2,
                                               OTParams* p, int level) {
  __shared__ unsigned lh[NBINS];
  for (int i = threadIdx.x; i < NBINS; i += 256) lh[i] = 0;
  __syncthreads();

  int lane = threadIdx.x & 31, wv = threadIdx.x >> 5;
  int tile = blockIdx.x * 8 + wv;
  int ti = tile >> 9, tj = tile & (TILES - 1);
  int lid = lane & 15, hl = lane >> 4, koff = hl * 2;
  v8f c = tile_dot(xp, xt, ti * 16, tj * 16, lid, koff);

  float lo, w;
  if (level == 0) {
    lo = __uint_as_float(p->minBits);
    w = (__uint_as_float(p->maxBits) - lo) / (float)NBINS;
  } else {
    lo = p->lo2;
    w = p->w2;
  }
  float invw = (w > 0.f) ? 1.0f / w : 0.0f;

  int j = tj * 16 + lid;
  float yj = y2[j];
#pragma unroll
  for (int r = 0; r < 8; ++r) {
    int i = ti * 16 + r + 8 * hl;
    float d2 = fmaxf(x2[i] + yj - 2.0f * c[r], 0.0f);
    int bin = (int)((d2 - lo) * invw);
    if (level == 0) bin = min(max(bin, 0), NBINS - 1);
    if (bin >= 0 && bin < NBINS) atomicAdd(&lh[bin], 1u);
  }
  __syncthreads();
  unsigned* gh = (level == 0) ? p->hist1 : p->hist2;
  for (int i = threadIdx.x; i < NBINS; i += 256)
    if (lh[i]) atomicAdd(&gh[i], lh[i]);
}

__global__ void ot_scan1(OTParams* p) {
  if (threadIdx.x != 0 || blockIdx.x != 0) return;
  float lo = __uint_as_float(p->minBits);
  float hi = __uint_as_float(p->maxBits);
  float w = (hi > lo) ? (hi - lo) / (float)NBINS : 0.0f;
  const unsigned long long target = ((unsigned long long)NPTS * NPTS) / 2;
  unsigned long long cum = 0, before = 0;
  int b = NBINS - 1;
  for (int i = 0; i < NBINS; ++i) {
    unsigned long long h = p->hist1[i];
    if (cum + h >= target) { b = i; before = cum; break; }
    cum += h;
    before = cum;
  }
  p->lo2 = lo + (float)b * w;
  p->w2 = w / (float)NBINS;
  p->below2 = before;
}

__global__ void ot_scan2(OTParams* p) {
  if (threadIdx.x != 0 || blockIdx.x != 0) return;
  const unsigned long long target = ((unsigned long long)NPTS * NPTS) / 2;
  unsigned long long cum = p->below2;
  int b = NBINS - 1;
  for (int i = 0; i < NBINS; ++i) {
    cum += p->hist2[i];
    if (cum >= target) { b = i; break; }
  }
  float med = (p->w2 > 0.f) ? (p->lo2 + ((float)b + 0.5f) * p->w2) : p->lo2;
  p->med = med;
  p->mscale = (med > 0.f) ? 1.0f / med : 1.0f;     // M normalization
  p->kscale = p->mscale * INV_REG;                  // exp argument scale
}

// K[i*N+j] = exp(-d2(i,j) * kscale) stored fp16. Called twice (xp,xt)->K and
// (xt,xp)->KT so both GEMV directions read row-major.
__global__ __launch_bounds__(256) void ot_buildk(const float* __restrict__ xp,
                                                 const float* __restrict__ xt,
                                                 const float* __restrict__ x2,
                                                 const float* __restrict__ y2,
                                                 const OTParams* __restrict__ p,
                                                 _Float16* __restrict__ Kout) {
  int lane = threadIdx.x & 31, wv = threadIdx.x >> 5;
  int tile = blockIdx.x * 8 + wv;
  int ti = tile >> 9, tj = tile & (TILES - 1);
  int lid = lane & 15, hl = lane >> 4, koff = hl * 2;
  v8f c = tile_dot(xp, xt, ti * 16, tj * 16, lid, koff);
  float ks = p->kscale;
  int j = tj * 16 + lid;
  float yj = y2[j];
#pragma unroll
  for (int r = 0; r < 8; ++r) {
    int i = ti * 16 + r + 8 * hl;
    float d2 = fmaxf(x2[i] + yj - 2.0f * c[r], 0.0f);
    Kout[(size_t)i * NPTS + j] = (_Float16)__expf(-d2 * ks);
  }
}

// out[row] = (1/N) / dot(A[row,:], x)   (one wave per row, x staged in LDS)
__global__ __launch_bounds__(256) void ot_gemv_div(const _Float16* __restrict__ A,
                                                   const float* __restrict__ x,
                                                   float* __restrict__ out) {
  __shared__ float xs[NPTS];
  for (int i = threadIdx.x; i < NPTS; i += 256) xs[i] = x[i];
  __syncthreads();
  int lane = threadIdx.x & 31, wv = threadIdx.x >> 5;
  int row = blockIdx.x * 8 + wv;
  const _Float16* ar = A + (size_t)row * NPTS;
  float acc = 0.f;
#pragma unroll 4
  for (int step = 0; step < NPTS / 256; ++step) {
    int col = step * 256 + lane * 8;
    h8 a = *(const h8*)(ar + col);   // 16B aligned: rows are 16KB apart
#pragma unroll
    for (int e = 0; e < 8; ++e) acc += (float)a[e] * xs[col + e];
  }
  acc = wave_sum(acc);
  if (lane == 0) out[row] = (1.0f / (float)NPTS) / acc;
}

// loss += u_i * exp(-M*10) * v_j * M, with M = d2 * mscale recomputed in fp32.
__global__ __launch_bounds__(256) void ot_loss(const float* __restrict__ xp,
                                               const float* __restrict__ xt,
                                               const float* __restrict__ x2,
                                               const float* __restrict__ y2,
                                               const float* __restrict__ u,
                                               const float* __restrict__ v,
                                               const OTParams* __restrict__ p,
                                               float* __restrict__ out) {
  int lane = threadIdx.x & 31, wv = threadIdx.x >> 5;
  int tile = blockIdx.x * 8 + wv;
  int ti = tile >> 9, tj = tile & (TILES - 1);
  int lid = lane & 15, hl = lane >> 4, koff = hl * 2;
  v8f c = tile_dot(xp, xt, ti * 16, tj * 16, lid, koff);
  float ms = p->mscale;
  int j = tj * 16 + lid;
  float yj = y2[j], vj = v[j];
  float acc = 0.f;
#pragma unroll
  for (int r = 0; r < 8; ++r) {
    int i = ti * 16 + r + 8 * hl;
    float d2 = fmaxf(x2[i] + yj - 2.0f * c[r], 0.0f);
    float m = d2 * ms;
    acc += u[i] * __expf(-m * INV_REG) * vj * m;
  }
  acc = wave_sum(acc);
  __shared__ float ls[8];
  if (lane == 0) ls[wv] = acc;
  __syncthreads();
  if (wv == 0) {
    float t = (lane < 8) ? ls[lane] : 0.f;
    t = wave_sum(t);
    if (lane == 0) atomicAdd(out, t);
  }
}

extern "C" void kernel_launch(void* const* d_in, const int* in_sizes, int n_in,
                              void* d_out, int out_size, void* d_ws, size_t ws_size,
                              hipStream_t stream) {
  const float* xp = (const float*)d_in[0];
  const float* xt = (const float*)d_in[1];
  float* out = (float*)d_out;

  char* ws = (char*)d_ws;
  const size_t KBYTES = (size_t)NPTS * NPTS * sizeof(_Float16);  // 128 MB
  _Float16* K  = (_Float16*)ws;
  _Float16* KT = (_Float16*)(ws + KBYTES);
  float* x2 = (float*)(ws + 2 * KBYTES);
  float* y2 = x2 + NPTS;
  float* u  = y2 + NPTS;
  float* v  = u + NPTS;
  OTParams* p = (OTParams*)(v + NPTS);

  const int TB = (TILES * TILES) / 8;   // tile-kernel blocks (8 waves/block)

  ot_init<<<32, 256, 0, stream>>>(p, u, v, out);
  ot_rownorm<<<(2 * NPTS) / 8, 256, 0, stream>>>(xp, xt, x2, y2);

  // median of M via 2-level histogram (tiles recomputed with WMMA each pass)
  ot_minmax<<<TB, 256, 0, stream>>>(xp, xt, x2, y2, p);
  ot_hist<<<TB, 256, 0, stream>>>(xp, xt, x2, y2, p, 0);
  ot_scan1<<<1, 32, 0, stream>>>(p);
  ot_hist<<<TB, 256, 0, stream>>>(xp, xt, x2, y2, p, 1);
  ot_scan2<<<1, 32, 0, stream>>>(p);

  // materialize K (row-major) and K^T (row-major, by swapping point sets)
  ot_buildk<<<TB, 256, 0, stream>>>(xp, xt, x2, y2, p, K);
  ot_buildk<<<TB, 256, 0, stream>>>(xt, xp, y2, x2, p, KT);

  // Sinkhorn iterations: v = nu / (K^T u); u = mu / (K v)
  for (int it = 0; it < NITER; ++it) {
    ot_gemv_div<<<NPTS / 8, 256, 0, stream>>>(KT, u, v);
    ot_gemv_div<<<NPTS / 8, 256, 0, stream>>>(K, v, u);
  }

  ot_loss<<<TB, 256, 0, stream>>>(xp, xt, x2, y2, u, v, p, out);
}